// GConv_13829794693475
// MI455X (gfx1250) — compile-verified
//
#include <hip/hip_runtime.h>

// GraphConv for MI455X (gfx1250, wave32).
//
// Phase 1: CSR build (int atomics + LDS scan) -> deterministic pull aggregation:
//          agg[n,0:256]   = sum_{e: dst[e]==n} feat[src[e]] * rsqrt(max(outdeg,1))
//          agg[n,256:320] = sum_{e: dst[e]==n} edge_feat[e]
//          (gathers are coalesced + L2-resident; no f32 atomics at all)
// Phase 2: out = relu((agg @ W) * rsqrt(max(indeg,1)) + bias)
//          via chained V_WMMA_F32_16X16X4_F32, A tile staged in LDS
//          (bank-conflict-free, 8x global-traffic reduction per block).

#define F_IN   256
#define F_E    64
#define F_OUT  256
#define K_DIM  320              // F_IN + F_E
#define K_PAD  324              // LDS row pitch: 324 mod 64 == 4 -> conflict-free b64 reads
#define SCAN_BD 256
#define SEG_CAP 1024

typedef __attribute__((ext_vector_type(2))) float v2f;
typedef __attribute__((ext_vector_type(8))) float v8f;

// ---------------------------------------------------------------- zero ints
__global__ void init_int_kernel(int* __restrict__ p, long total) {
    long i = (long)blockIdx.x * blockDim.x + threadIdx.x;
    long stride = (long)gridDim.x * blockDim.x;
    for (; i < total; i += stride) p[i] = 0;
}

// ------------------------------------------------------------- degree count
__global__ void degree_kernel(const int* __restrict__ src, const int* __restrict__ dst,
                              int* __restrict__ cnt_out, int* __restrict__ cnt_in, int E) {
    int e = blockIdx.x * blockDim.x + threadIdx.x;
    if (e < E) {
        atomicAdd(&cnt_out[src[e]], 1);
        atomicAdd(&cnt_in[dst[e]], 1);
    }
}

// -------------------------------------------------- degree -> rsqrt(max(d,1))
__global__ void norm_kernel(const int* __restrict__ cnt_out, const int* __restrict__ cnt_in,
                            float* __restrict__ norm_out, float* __restrict__ norm_in, int N) {
    int i = blockIdx.x * blockDim.x + threadIdx.x;
    if (i < N) {
        norm_out[i] = rsqrtf(fmaxf((float)cnt_out[i], 1.0f));
        norm_in[i]  = rsqrtf(fmaxf((float)cnt_in[i],  1.0f));
    }
}

// ---------------------------------------------------- two-level scan (CSR offs)
__global__ void scan_blocks_kernel(const int* __restrict__ cnt, int* __restrict__ partial,
                                   int* __restrict__ bsum, int N) {
    __shared__ int sh[SCAN_BD];
    int gid = blockIdx.x * SCAN_BD + threadIdx.x;
    int v = (gid < N) ? cnt[gid] : 0;
    sh[threadIdx.x] = v;
    __syncthreads();
    for (int off = 1; off < SCAN_BD; off <<= 1) {
        int t = (threadIdx.x >= off) ? sh[threadIdx.x - off] : 0;
        __syncthreads();
        sh[threadIdx.x] += t;
        __syncthreads();
    }
    if (gid < N) partial[gid] = sh[threadIdx.x];          // inclusive within block
    if (threadIdx.x == SCAN_BD - 1) bsum[blockIdx.x] = sh[SCAN_BD - 1];
}

// single block; serial-carry scan of block sums -> exclusive block offsets
__global__ void scan_tops_kernel(int* __restrict__ bsum, int C) {
    __shared__ int sh[SCAN_BD];
    __shared__ int carry_sh;
    if (threadIdx.x == 0) carry_sh = 0;
    __syncthreads();
    for (int base = 0; base < C; base += SCAN_BD) {
        int i = base + threadIdx.x;
        int v = (i < C) ? bsum[i] : 0;
        sh[threadIdx.x] = v;
        __syncthreads();
        for (int off = 1; off < SCAN_BD; off <<= 1) {
            int t = (threadIdx.x >= off) ? sh[threadIdx.x - off] : 0;
            __syncthreads();
            sh[threadIdx.x] += t;
            __syncthreads();
        }
        int incl = sh[threadIdx.x];
        int carry = carry_sh;
        __syncthreads();
        if (i < C) bsum[i] = carry + incl - v;            // exclusive
        if (threadIdx.x == SCAN_BD - 1) carry_sh = carry + incl;
        __syncthreads();
    }
}

__global__ void scan_finalize_kernel(const int* __restrict__ partial, const int* __restrict__ cnt,
                                     const int* __restrict__ bsum, int* __restrict__ offs,
                                     int N, int E) {
    int gid = blockIdx.x * SCAN_BD + threadIdx.x;
    if (gid < N) offs[gid] = partial[gid] - cnt[gid] + bsum[blockIdx.x];  // global exclusive
    if (gid == 0) offs[N] = E;
}

// ------------------------------------------------------------- CSR edge fill
__global__ void fill_kernel(const int* __restrict__ dst, const int* __restrict__ offs,
                            int* __restrict__ cursor, int* __restrict__ eidx, int E) {
    int e = blockIdx.x * blockDim.x + threadIdx.x;
    if (e < E) {
        int d = dst[e];
        int p = atomicAdd(&cursor[d], 1);
        eidx[offs[d] + p] = e;
    }
}

// -------------------------------------------- pull aggregation (deterministic)
// One block (320 threads) per dst node. Segment edge ids staged + sorted in LDS
// so f32 accumulation order is input-determined (bitwise-stable across calls).
__global__ void aggregate_kernel(const float* __restrict__ feat,
                                 const float* __restrict__ edge_feat,
                                 const int* __restrict__ src,
                                 const int* __restrict__ eidx,
                                 const int* __restrict__ offs,
                                 const float* __restrict__ norm_out,
                                 float* __restrict__ agg) {
    __shared__ int seg[SEG_CAP];
    const int n = blockIdx.x;
    const int tid = threadIdx.x;
    const int beg = offs[n];
    const int len = offs[n + 1] - beg;
    const bool fits = (len <= SEG_CAP);

    for (int j = tid; j < len && j < SEG_CAP; j += blockDim.x) seg[j] = eidx[beg + j];
    __syncthreads();

    if (fits) {   // odd-even transposition sort (len ~ 16)
        for (int pass = 0; pass < len; ++pass) {
            for (int j = (pass & 1) + 2 * tid; j + 1 < len; j += 2 * blockDim.x) {
                int a = seg[j], b = seg[j + 1];
                if (a > b) { seg[j] = b; seg[j + 1] = a; }
            }
            __syncthreads();
        }
    }

    float acc = 0.0f;
    if (tid < F_IN) {
        for (int j = 0; j < len; ++j) {
            int e = fits ? seg[j] : eidx[beg + j];
            int s = src[e];                                   // broadcast load
            if (j + 1 < len) {                                // prefetch next row
                int e2 = fits ? seg[j + 1] : eidx[beg + j + 1];
                __builtin_prefetch(&feat[(size_t)src[e2] * F_IN + tid], 0, 0);
            }
            acc += feat[(size_t)s * F_IN + tid] * norm_out[s];
        }
        agg[(size_t)n * K_DIM + tid] = acc;
    } else {
        int c = tid - F_IN;
        for (int j = 0; j < len; ++j) {
            int e = fits ? seg[j] : eidx[beg + j];
            acc += edge_feat[(size_t)e * F_E + c];
        }
        agg[(size_t)n * K_DIM + F_IN + c] = acc;
    }
}

// ------------------------------------------------------------ WMMA GEMM + epilogue
// Block = 256 threads = 8 waves; all waves share one 16-row A tile staged in LDS
// (pitch K_PAD=324 -> ds_load_b64 A reads hit all 64 banks exactly once).
// Wave w owns cols [by*128 + w*16, +16); 80 chained V_WMMA_F32_16X16X4_F32.
__global__ void gemm_kernel(const float* __restrict__ agg,
                            const float* __restrict__ W,
                            const float* __restrict__ bias,
                            const float* __restrict__ norm_in,
                            float* __restrict__ out, int N) {
    __shared__ float Atile[16 * K_PAD];   // ~20.7 KB
    __shared__ float ntile[16];

    const int tid  = threadIdx.x;
    const int lane = tid & 31;
    const int wave = tid >> 5;
    const int half = lane >> 4;
    const int l15  = lane & 15;
    const int m0   = blockIdx.x * 16;
    const int n0   = blockIdx.y * 128 + wave * 16;

    int rows = N - m0; if (rows > 16) rows = 16;

    // cooperative copy: tile is contiguous in global ([m0*320, (m0+rows)*320))
    const float4* __restrict__ gsrc = (const float4*)(agg + (size_t)m0 * K_DIM);
    float4* lds4 = (float4*)Atile;                       // K_PAD/4 = 81 float4 pitch
    for (int i = tid; i < rows * (K_DIM / 4); i += 256) {
        int r = i / (K_DIM / 4);
        int c = i - r * (K_DIM / 4);
        lds4[r * (K_PAD / 4) + c] = gsrc[i];
    }
    if (tid < rows) ntile[tid] = norm_in[m0 + tid];
    __syncthreads();

    const float* __restrict__ arow = Atile + l15 * K_PAD + half * 2;
    const float* __restrict__ bcol = W + (size_t)(half * 2) * F_OUT + n0 + l15;

    v8f c = {};
    for (int kk = 0; kk < K_DIM; kk += 4) {
        v2f a, b;
        a.x = arow[kk];                 // ds_load_b64, conflict-free
        a.y = arow[kk + 1];
        b.x = bcol[(size_t)kk * F_OUT];
        b.y = bcol[(size_t)kk * F_OUT + F_OUT];
        // (neg_a, A, neg_b, B, c_mod, C, reuse_a, reuse_b)
        c = __builtin_amdgcn_wmma_f32_16x16x4_f32(false, a, false, b,
                                                  (short)0, c, false, false);
    }

    const float bval = bias[n0 + l15];
    if (rows == 16) {                                    // uniform fast path
#pragma unroll
        for (int r = 0; r < 8; ++r) {
            int m = m0 + r + half * 8;
            float v = c[r] * ntile[r + half * 8] + bval;
            out[(size_t)m * F_OUT + n0 + l15] = fmaxf(v, 0.0f);
        }
    } else {
#pragma unroll
        for (int r = 0; r < 8; ++r) {
            int m = m0 + r + half * 8;
            if (m < N) {
                float v = c[r] * ntile[r + half * 8] + bval;
                out[(size_t)m * F_OUT + n0 + l15] = fmaxf(v, 0.0f);
            }
        }
    }
}

// ------------------------------------------------------------------ launcher
extern "C" void kernel_launch(void* const* d_in, const int* in_sizes, int n_in,
                              void* d_out, int out_size, void* d_ws, size_t ws_size,
                              hipStream_t stream) {
    const float* feat      = (const float*)d_in[0];
    const float* edge_feat = (const float*)d_in[1];
    const float* weight    = (const float*)d_in[2];
    const float* bias      = (const float*)d_in[3];
    const int*   src       = (const int*)d_in[4];
    const int*   dst       = (const int*)d_in[5];

    const int N = in_sizes[0] / F_IN;   // 50000
    const int E = in_sizes[4];          // 800000
    const int C = (N + SCAN_BD - 1) / SCAN_BD;

    // workspace layout (all 4-byte types)
    float* agg      = (float*)d_ws;                 // N*320 floats (~64 MB)
    int*   cnt_out  = (int*)(agg + (size_t)N * K_DIM);  // N
    int*   cnt_in   = cnt_out + N;                  // N
    int*   cursor   = cnt_in + N;                   // N   (contiguous w/ cnt_* for init)
    int*   offs     = cursor + N;                   // N+1
    int*   partial  = offs + N + 1;                 // N
    int*   bsum     = partial + N;                  // C
    float* norm_out = (float*)(bsum + C);           // N
    float* norm_in  = norm_out + N;                 // N
    int*   eidx     = (int*)(norm_in + N);          // E

    init_int_kernel<<<512, 256, 0, stream>>>(cnt_out, 3L * N);   // cnt_out,cnt_in,cursor

    degree_kernel<<<(E + 255) / 256, 256, 0, stream>>>(src, dst, cnt_out, cnt_in, E);

    norm_kernel<<<(N + 255) / 256, 256, 0, stream>>>(cnt_out, cnt_in, norm_out, norm_in, N);

    scan_blocks_kernel<<<C, SCAN_BD, 0, stream>>>(cnt_in, partial, bsum, N);
    scan_tops_kernel<<<1, SCAN_BD, 0, stream>>>(bsum, C);
    scan_finalize_kernel<<<C, SCAN_BD, 0, stream>>>(partial, cnt_in, bsum, offs, N, E);

    fill_kernel<<<(E + 255) / 256, 256, 0, stream>>>(dst, offs, cursor, eidx, E);

    aggregate_kernel<<<N, K_DIM, 0, stream>>>(feat, edge_feat, src, eidx, offs,
                                              norm_out, agg);

    dim3 grid((N + 15) / 16, F_OUT / 128);
    gemm_kernel<<<grid, 256, 0, stream>>>(agg, weight, bias, norm_in, (float*)d_out, N);
}